// DF_89653147336831
// MI455X (gfx1250) — compile-verified
//
#include <hip/hip_runtime.h>
#include <hip/hip_bf16.h>
#include <math.h>

#define NUM_BIN     257
#define NUM_CH      8
#define NUM_BLOCK   36
#define NUM_FRAME   400
#define FRAMES_PER_TILE 16
#define BINS_PER_CHUNK  32
#define FSTRIDE 20                              // dwords per frame row (16 comps + 4 pad)
#define BSTRIDE (FRAMES_PER_TILE * FSTRIDE + 8) // 328 dwords per bin (8 mod 64, 16B aligned)
#define OSTRIDE 33                              // out_buf row stride (odd -> bank spread)

typedef __attribute__((ext_vector_type(2))) float v2f;
typedef __attribute__((ext_vector_type(8))) float v8f;

// DF(b,f,bin) = (1/28) * x^T G x,  x = [cos(ang_c), sin(ang_c)]_{c=0..7}.
// Transposed WMMA form: D' = G^T (A-operand) x X^T (B-operand) via 4 chained
// v_wmma_f32_16x16x4_f32; lane n then owns frame n's column of D', so the
// final dot is 8 in-lane FMAs + a single shfl_xor(16) to merge comp halves.
// G-entry selectors (pair validity, ct value, sign, cos/sin) are per-lane
// constants hoisted out of all loops -> branch-free per-bin G build.
__global__ __launch_bounds__(256) void df_wmma_kernel(
    const float* __restrict__ inp,      // (36*400, 2, 257, 8)
    const float* __restrict__ tdoas,    // (36,)
    const float* __restrict__ ctheta,   // (8,8)
    float* __restrict__ out)            // (36, 400, 257)
{
  __shared__ float lds[BINS_PER_CHUNK * BSTRIDE];           // ~41 KB
  __shared__ float out_buf[FRAMES_PER_TILE * OSTRIDE];      // 2.1 KB

  const int tid  = threadIdx.x;
  const int lane = tid & 31;
  const int wave = tid >> 5;
  const int ft   = blockIdx.x;        // frame tile 0..24
  const int b    = blockIdx.y;        // block 0..35

  const float cd     = cosf(tdoas[b] * (1.0f / 180.0f));
  const float fscale = 8000.0f / (float)(NUM_BIN - 1);
  const int   f0     = b * NUM_FRAME + ft * FRAMES_PER_TILE;  // first global frame

  const int n  = lane & 15;   // matrix column index (frame within tile)
  const int hi = lane >> 4;   // half-wave select

  // ---- Hoisted per-lane G-entry constants (bin-independent) ----
  // Entry (t,j): row k = 4t+2*hi+j of G, column n.  p = k&7, q = n&7.
  // G[k][n] = valid(p<q) * { cos block: +cos(v); s-row/c-col: +sin(v);
  //                          c-row/s-col: -sin(v) },  v = theta*ct[p][q].
  float ctv[8];   // ct[p][q], 0 when entry unused
  float cf[8];    // 0, +1, -1 coefficient
  bool  iscos[8]; // pick cos vs sin
  #pragma unroll
  for (int t = 0; t < 4; ++t) {
    #pragma unroll
    for (int j = 0; j < 2; ++j) {
      const int idx = 2 * t + j;
      const int k = 4 * t + 2 * hi + j;
      const int p = k & 7, q = n & 7;
      const bool valid = (p < q);
      const bool ks = (k >= 8), ns = (n >= 8);
      iscos[idx] = (ks == ns);
      cf[idx]    = valid ? ((ks == ns) ? 1.0f : (ks ? 1.0f : -1.0f)) : 0.0f;
      ctv[idx]   = valid ? ctheta[p * 8 + q] : 0.0f;
    }
  }

  for (int bin0 = 0; bin0 < NUM_BIN; bin0 += BINS_PER_CHUNK) {
    const int nb = min(BINS_PER_CHUNK, NUM_BIN - bin0);
    __syncthreads();   // protect lds reuse across chunk iterations

    // ---- Phase 1: b128 global loads, rsqrt-normalize, b128 LDS stores ----
    {
      const int fl  = tid >> 4;     // local frame 0..15
      const int l16 = tid & 15;
      const float* fre = inp + ((size_t)(f0 + fl) * 2 + 0) * (NUM_BIN * NUM_CH)
                             + (size_t)bin0 * NUM_CH;
      const float* fim = fre + (NUM_BIN * NUM_CH);
      for (int e4 = l16; e4 < nb * 2; e4 += 16) {
        float4 re4 = *(const float4*)(fre + e4 * 4);
        float4 im4 = *(const float4*)(fim + e4 * 4);
        float4 c4, s4;
        {
          float r;
          r = rsqrtf(fmaxf(re4.x*re4.x + im4.x*im4.x, 1e-30f)); c4.x = re4.x*r; s4.x = im4.x*r;
          r = rsqrtf(fmaxf(re4.y*re4.y + im4.y*im4.y, 1e-30f)); c4.y = re4.y*r; s4.y = im4.y*r;
          r = rsqrtf(fmaxf(re4.z*re4.z + im4.z*im4.z, 1e-30f)); c4.z = re4.z*r; s4.z = im4.z*r;
          r = rsqrtf(fmaxf(re4.w*re4.w + im4.w*im4.w, 1e-30f)); c4.w = re4.w*r; s4.w = im4.w*r;
        }
        const int bl  = e4 >> 1;           // bin within chunk
        const int off = (e4 & 1) * 4;      // channel group 0..3 / 4..7
        float* dst = &lds[bl * BSTRIDE + fl * FSTRIDE + off];
        *(float4*)(dst)     = c4;          // cos comps 0..7
        *(float4*)(dst + 8) = s4;          // sin comps 8..15
      }
    }
    __syncthreads();

    // ---- Phase 2: per-wave WMMA over bins (uniform trip count per wave) ----
    for (int bl = wave; bl < nb; bl += 8) {
      const int   bing  = bin0 + bl;
      const float theta = fscale * (float)bing * cd;

      // Branch-free G^T build in A-operand layout (EXEC stays all-ones).
      v2f bm[4];
      #pragma unroll
      for (int t = 0; t < 4; ++t) {
        float ej[2];
        #pragma unroll
        for (int j = 0; j < 2; ++j) {
          const int idx = 2 * t + j;
          float sv, cv;
          __sincosf(theta * ctv[idx], &sv, &cv);
          ej[j] = cf[idx] * (iscos[idx] ? cv : sv);
        }
        v2f bt; bt[0] = ej[0]; bt[1] = ej[1];
        bm[t] = bt;
      }

      // X^T in B-operand layout: lane col = frame n, VGPRs = comps 4t+2*hi+j.
      const float* xrow = &lds[bl * BSTRIDE + n * FSTRIDE];
      v2f am[4];
      #pragma unroll
      for (int t = 0; t < 4; ++t)
        am[t] = *(const v2f*)(xrow + 4 * t + 2 * hi);

      // This lane's x slice in D layout: comps 8*hi .. 8*hi+7, contiguous.
      const float4 xj0 = *(const float4*)(xrow + 8 * hi);
      const float4 xj1 = *(const float4*)(xrow + 8 * hi + 4);

      // D' = G^T * X^T via 4 chained f32 WMMAs (K = 16).
      v8f acc = {};
      #pragma unroll
      for (int t = 0; t < 4; ++t)
        acc = __builtin_amdgcn_wmma_f32_16x16x4_f32(
            false, bm[t], false, am[t], (short)0, acc, false, false);

      // P[frame n] = sum over 16 comps: 8 in-lane FMAs + one cross-half add.
      float pv = acc[0]*xj0.x + acc[1]*xj0.y + acc[2]*xj0.z + acc[3]*xj0.w
               + acc[4]*xj1.x + acc[5]*xj1.y + acc[6]*xj1.z + acc[7]*xj1.w;
      pv += __shfl_xor(pv, 16);
      if (hi == 0)
        out_buf[n * OSTRIDE + bl] = pv * (1.0f / 28.0f);
    }
    __syncthreads();

    // ---- Phase 3: coalesced writeback of the 16 x nb result tile ----
    {
      const int fr = tid >> 4;
      const int c0 = tid & 15;
      const size_t base = (size_t)(f0 + fr) * NUM_BIN + bin0;
      for (int c = c0; c < nb; c += 16)
        out[base + c] = out_buf[fr * OSTRIDE + c];
    }
  }
}

extern "C" void kernel_launch(void* const* d_in, const int* in_sizes, int n_in,
                              void* d_out, int out_size, void* d_ws, size_t ws_size,
                              hipStream_t stream) {
  (void)in_sizes; (void)n_in; (void)d_ws; (void)ws_size; (void)out_size;
  const float* inp = (const float*)d_in[0];   // input  (14400,2,257,8) f32
  const float* tdo = (const float*)d_in[1];   // tdoas  (36,)           f32
  const float* ct  = (const float*)d_in[2];   // const_theta (8,8)      f32
  // d_in[3] = pair_id (28,2) int32 — pair enumeration i<j is hardcoded.
  float* out = (float*)d_out;                 // (36,400,257) f32

  dim3 grid(NUM_FRAME / FRAMES_PER_TILE, NUM_BLOCK);  // 25 x 36 workgroups
  df_wmma_kernel<<<grid, 256, 0, stream>>>(inp, tdo, ct, out);
}